// HODLRLoss_27891517621092
// MI455X (gfx1250) — compile-verified
//
#include <hip/hip_runtime.h>

typedef float v2f __attribute__((ext_vector_type(2)));
typedef float v8f __attribute__((ext_vector_type(8)));

#define NTOT   262144
#define LEAF   32
#define NB     (NTOT / LEAF)   // 8192 leaf blocks
#define RANK   16
#define NLEV   8
#define NNZ_   1310720
#define TSEGS  510             // sum_{l=0..7} 2^(l+1)

struct Ptr8 { const float* p[8]; };

// ---------------------------------------------------------------------------
// K0: zero the regions that are accumulated into (y, t, loss)
// ---------------------------------------------------------------------------
__global__ void zero_kernel(float* __restrict__ p, int n) {
    int i = blockIdx.x * blockDim.x + threadIdx.x;
    if (i < n) p[i] = 0.0f;
}

// ---------------------------------------------------------------------------
// K1: per-leaf  w = L * (L^T * x) + 1e-4 * x
// Strategy: build M = L*L^T (four 16x16 tiles) with v_wmma_f32_16x16x4_f32,
// K=32 split into 8 chunks of 4. With the 16x16x4 fragment layouts, the
// B-fragment of L^T (B[k][j] = L[16q+j][4kc+k]) is element-identical per lane
// to the A-fragment of L tile q, so each k-chunk needs only two loads.
// Then matvec M*x from LDS. One wave (32 lanes) per leaf; EXEC all-ones.
// ---------------------------------------------------------------------------
__global__ __launch_bounds__(32) void leaf_kernel(const float* __restrict__ Lg,
                                                  const float* __restrict__ z,
                                                  float* __restrict__ w) {
    __shared__ float Ls[32][33];
    __shared__ float Ms[32][33];
    __shared__ float xs[32];

    const int b = blockIdx.x;
    const int t = threadIdx.x;
    const float* Lb = Lg + (size_t)b * (LEAF * LEAF);

    // coalesced load of L (row-major 32x32) into LDS
    for (int j = t; j < LEAF * LEAF; j += 32)
        Ls[j >> 5][j & 31] = Lb[j];
    xs[t] = z[b * LEAF + t];
    __syncthreads();

    const int tl = t & 15;       // column / row-in-tile
    const int th = t >> 4;       // half-wave selector (K pairing)

    v8f c00 = {}, c01 = {}, c10 = {}, c11 = {};

    #pragma unroll
    for (int kc = 0; kc < 8; ++kc) {
        const int k0 = kc * 4 + th * 2;
        // A-fragment of tile p (16x4): lane(t): rows tl (+16 for p=1),
        // VGPR0=K(k0), VGPR1=K(k0+1). Identical layout serves as B-fragment
        // of (tile q)^T because M = L*L^T is symmetric in the K index.
        v2f a0, a1;
        a0.x = Ls[tl][k0];        a0.y = Ls[tl][k0 + 1];
        a1.x = Ls[16 + tl][k0];   a1.y = Ls[16 + tl][k0 + 1];

        c00 = __builtin_amdgcn_wmma_f32_16x16x4_f32(false, a0, false, a0, (short)0, c00, false, false);
        c01 = __builtin_amdgcn_wmma_f32_16x16x4_f32(false, a0, false, a1, (short)0, c01, false, false);
        c10 = __builtin_amdgcn_wmma_f32_16x16x4_f32(false, a1, false, a0, (short)0, c10, false, false);
        c11 = __builtin_amdgcn_wmma_f32_16x16x4_f32(false, a1, false, a1, (short)0, c11, false, false);
    }

    // C/D layout: VGPR v -> row v (lanes 0-15) or row v+8 (lanes 16-31), col = tl
    #pragma unroll
    for (int v = 0; v < 8; ++v) {
        const int r = v + 8 * th;
        Ms[r][tl]           = c00[v];
        Ms[r][16 + tl]      = c01[v];
        Ms[16 + r][tl]      = c10[v];
        Ms[16 + r][16 + tl] = c11[v];
    }
    __syncthreads();

    // y[t] = sum_j M[t][j] * x[j] + 1e-4 * x[t]
    float acc = 1.0e-4f * xs[t];
    #pragma unroll
    for (int j = 0; j < 32; ++j)
        acc += Ms[t][j] * xs[j];
    w[b * LEAF + t] = acc;
}

// ---------------------------------------------------------------------------
// K2: t[l][s][r] = sum_{i in segment s of level l} v_l[i][r] * z[i]
// Grid: x = 256 chunks of 1024 elements, y = level. A 1024-aligned chunk lies
// inside a single segment at every level (min bs = 1024), so each block does
// one in-LDS tree reduction and 16 global atomics.
// Thread t: r = t&15, group g = t>>4 ; each group covers 64 elements.
// ---------------------------------------------------------------------------
__global__ __launch_bounds__(256) void level_reduce_kernel(Ptr8 V,
                                                           const float* __restrict__ z,
                                                           float* __restrict__ tbuf) {
    __shared__ float xs[1024];
    __shared__ float part[16][17];

    const int l    = blockIdx.y;
    const int base = blockIdx.x * 1024;
    const int t    = threadIdx.x;
    const int r    = t & 15;
    const int g    = t >> 4;

    for (int j = t; j < 1024; j += 256)
        xs[j] = z[base + j];
    __syncthreads();

    const float* vl = V.p[l] + (size_t)base * RANK + g * RANK + r;
    float acc = 0.0f;
    #pragma unroll 4
    for (int e = 0; e < 64; ++e)
        acc += vl[e * 256] * xs[e * 16 + g];

    part[g][r] = acc;
    __syncthreads();

    if (t < 16) {
        float sum = 0.0f;
        #pragma unroll
        for (int gg = 0; gg < 16; ++gg)
            sum += part[gg][t];
        const int s = base >> (17 - l);              // bs_l = 2^(17-l)
        const int off = ((2 << l) - 2 + s) * RANK;   // level base + segment
        atomicAdd(&tbuf[off + t], sum);
    }
}

// ---------------------------------------------------------------------------
// K3: w[i] += sum_l sum_r u_l[i][r] * t[l][sibling(seg_l(i))][r]
// Coalesced float4 streaming of the u arrays (512 B / element total).
// All threads of a 256-block share the same segment at every level.
// ---------------------------------------------------------------------------
__global__ __launch_bounds__(256) void apply_kernel(Ptr8 U,
                                                    const float* __restrict__ tbuf,
                                                    float* __restrict__ w) {
    const int i = blockIdx.x * 256 + threadIdx.x;
    float acc = 0.0f;
    #pragma unroll
    for (int l = 0; l < NLEV; ++l) {
        const int s   = i >> (17 - l);
        const int off = ((2 << l) - 2 + (s ^ 1)) * RANK;
        const float4* u4 = (const float4*)(U.p[l] + (size_t)i * RANK);
        const float4* t4 = (const float4*)(tbuf + off);
        #pragma unroll
        for (int q = 0; q < 4; ++q) {
            float4 u = u4[q], tv = t4[q];
            acc += u.x * tv.x + u.y * tv.y + u.z * tv.z + u.w * tv.w;
        }
    }
    w[i] += acc;
}

// ---------------------------------------------------------------------------
// K4: sparse scatter  y[row] += val * w[col]   (A_indices is int64, [2,NNZ])
// ---------------------------------------------------------------------------
__global__ __launch_bounds__(256) void spmv_kernel(const long long* __restrict__ rows,
                                                   const long long* __restrict__ cols,
                                                   const float* __restrict__ vals,
                                                   const float* __restrict__ w,
                                                   float* __restrict__ y) {
    const int k = blockIdx.x * 256 + threadIdx.x;
    if (k < NNZ_) {
        const int rr = (int)rows[k];
        const int cc = (int)cols[k];
        atomicAdd(&y[rr], vals[k] * w[cc]);
    }
}

// ---------------------------------------------------------------------------
// K5: loss partial sums:  lossbuf += sum (y[i]-z[i])^2
// ---------------------------------------------------------------------------
__global__ __launch_bounds__(256) void loss_kernel(const float* __restrict__ y,
                                                   const float* __restrict__ z,
                                                   float* __restrict__ lossbuf) {
    __shared__ float s[8];
    const int i = blockIdx.x * 256 + threadIdx.x;
    const float d = y[i] - z[i];
    float v = d * d;
    #pragma unroll
    for (int off = 16; off > 0; off >>= 1)
        v += __shfl_down(v, off, 32);
    const int lane = threadIdx.x & 31, wv = threadIdx.x >> 5;
    if (lane == 0) s[wv] = v;
    __syncthreads();
    if (threadIdx.x == 0) {
        float sum = 0.0f;
        #pragma unroll
        for (int k = 0; k < 8; ++k) sum += s[k];
        atomicAdd(lossbuf, sum);
    }
}

__global__ void final_kernel(const float* __restrict__ lossbuf, float* __restrict__ out) {
    out[0] = lossbuf[0] * (1.0f / (float)NTOT);
}

// ---------------------------------------------------------------------------
extern "C" void kernel_launch(void* const* d_in, const int* in_sizes, int n_in,
                              void* d_out, int out_size, void* d_ws, size_t ws_size,
                              hipStream_t stream) {
    (void)in_sizes; (void)n_in; (void)out_size; (void)ws_size;

    const float* L = (const float*)d_in[0];
    Ptr8 U, V;
    for (int l = 0; l < NLEV; ++l) {
        U.p[l] = (const float*)d_in[1 + 2 * l];
        V.p[l] = (const float*)d_in[2 + 2 * l];
    }
    const long long* rows = (const long long*)d_in[17];
    const long long* cols = rows + NNZ_;
    const float* vals = (const float*)d_in[18];
    const float* z    = (const float*)d_in[19];

    float* w       = (float*)d_ws;             // [0, N)
    float* y       = w + NTOT;                 // [N, 2N)
    float* tbuf    = w + 2 * NTOT;             // 510*16 floats
    float* lossbuf = tbuf + TSEGS * RANK;      // 1 float

    const int nzero = NTOT + TSEGS * RANK + 1; // y, tbuf, lossbuf contiguous
    zero_kernel<<<(nzero + 255) / 256, 256, 0, stream>>>(y, nzero);

    leaf_kernel<<<NB, 32, 0, stream>>>(L, z, w);
    level_reduce_kernel<<<dim3(NTOT / 1024, NLEV), 256, 0, stream>>>(V, z, tbuf);
    apply_kernel<<<NTOT / 256, 256, 0, stream>>>(U, tbuf, w);
    spmv_kernel<<<(NNZ_ + 255) / 256, 256, 0, stream>>>(rows, cols, vals, w, y);
    loss_kernel<<<NTOT / 256, 256, 0, stream>>>(y, z, lossbuf);
    final_kernel<<<1, 1, 0, stream>>>(lossbuf, (float*)d_out);
}